// DistributedRealSHT_26448408609075
// MI455X (gfx1250) — compile-verified
//
#include <hip/hip_runtime.h>
#include <math.h>

typedef __attribute__((ext_vector_type(2))) float vf2;
typedef __attribute__((ext_vector_type(8))) float vf8;
typedef __attribute__((ext_vector_type(4))) unsigned int vu4;
typedef __attribute__((ext_vector_type(8))) int vi8;
typedef __attribute__((ext_vector_type(4))) int vi4;

#define NLON 1024
#define NLAT 512
#define BCDIM 64
#define MMAX 513
#define MP 528                    // 33 tiles of 16 covering 513 modes
#define RROWS (BCDIM * NLAT)      // 32768 combined (b,c,k) rows
#define TABN (MP * NLON)          // twiddle table entries per (cos|sin)

#define TROW 66                   // TDM-padded LDS row stride in dwords (64 + 2)
#define TILE_DW (16 * TROW)       // dwords per double-buffer slot

// ---------------------------------------------------------------------------
// Stage 1: exact DFT twiddle tables, m-major so WMMA B fragments are
// contiguous float2 loads.  fr[m][n] =  cos(2*pi*m*n/N) * (2*pi/N)
//                           fi[m][n] = -sin(2*pi*m*n/N) * (2*pi/N)
// ---------------------------------------------------------------------------
__global__ void sht_build_dft(float* __restrict__ fr, float* __restrict__ fi) {
    int idx = blockIdx.x * blockDim.x + threadIdx.x;
    if (idx >= TABN) return;
    int m = idx >> 10;
    int n = idx & 1023;
    int r = (m * n) & 1023;                 // exact phase mod N (N=1024)
    const float scale = 6.28318530717958647692f / (float)NLON;  // 2*pi/N
    float th = scale * (float)r;
    float s, c;
    __sincosf(th, &s, &c);
    fr[idx] = c * scale;
    fi[idx] = -s * scale;
}

// ---------------------------------------------------------------------------
// Stage 2: DFT as GEMM.  One wave per 32x16 output super-tile (two 16-row
// tiles sharing the cos/sin B fragments -> 4 loads per 4 WMMAs).
// Output stored transposed as X[m][row] so stage 3's A loads are contiguous.
// ---------------------------------------------------------------------------
__global__ void sht_dft_gemm(const float* __restrict__ x,
                             const float* __restrict__ fr,
                             const float* __restrict__ fi,
                             float* __restrict__ xr,
                             float* __restrict__ xi) {
    int wid  = blockIdx.x * (blockDim.x >> 5) + (threadIdx.x >> 5);
    int lane = threadIdx.x & 31;
    int rt = wid & 1023;          // 1024 row super-tiles (32 rows each)
    int mt = wid >> 10;           // 33 mode tiles
    int row0 = rt << 5;
    int m0   = mt << 4;
    int l16  = lane & 15;
    int kb   = (lane >> 4) << 1;  // 0 or 2

    const float* ap0 = x  + (size_t)(row0 + l16) * NLON + kb;
    const float* ap1 = ap0 + (size_t)16 * NLON;
    const float* brp = fr + (size_t)(m0 + l16) * NLON + kb;
    const float* bip = fi + (size_t)(m0 + l16) * NLON + kb;

    vf8 ar0 = {}, ai0 = {}, ar1 = {}, ai1 = {};
    for (int n = 0; n < NLON; n += 4) {
        vf2 a0 = *(const vf2*)(ap0 + n);
        vf2 a1 = *(const vf2*)(ap1 + n);
        vf2 br = *(const vf2*)(brp + n);
        vf2 bi = *(const vf2*)(bip + n);
        ar0 = __builtin_amdgcn_wmma_f32_16x16x4_f32(false, a0, false, br,
                                                    (short)0, ar0, false, false);
        ai0 = __builtin_amdgcn_wmma_f32_16x16x4_f32(false, a0, false, bi,
                                                    (short)0, ai0, false, false);
        ar1 = __builtin_amdgcn_wmma_f32_16x16x4_f32(false, a1, false, br,
                                                    (short)0, ar1, false, false);
        ai1 = __builtin_amdgcn_wmma_f32_16x16x4_f32(false, a1, false, bi,
                                                    (short)0, ai1, false, false);
    }

    int m = m0 + l16;
    if (m < MMAX) {
        int rbase = row0 + ((lane >> 4) << 3);   // +8 rows for upper half-wave
        float* pr = xr + (size_t)m * RROWS + rbase;
        float* pi = xi + (size_t)m * RROWS + rbase;
        #pragma unroll
        for (int v = 0; v < 8; ++v) {
            pr[v]      = ar0[v];
            pi[v]      = ai0[v];
            pr[v + 16] = ar1[v];
            pi[v + 16] = ai1[v];
        }
    }
}

// ---------------------------------------------------------------------------
// TDM descriptor issue: DMA a 16(l) x 64(k) f32 weights tile into LDS with
// 2-dword padding every 64-dword row (LDS row stride 66 dwords -> B-fragment
// reads hit distinct banks).  D# layout per CDNA5 ISA 8.3/8.4.
// ---------------------------------------------------------------------------
__device__ __forceinline__ void tdm_load_tile(unsigned lds_addr,
                                              unsigned long long gaddr) {
    vu4 g0;
    g0[0] = 1u;                                             // count=1, user mode
    g0[1] = lds_addr;                                       // LDS byte address
    g0[2] = (unsigned)(gaddr & 0xFFFFFFFFull);              // global_addr[31:0]
    g0[3] = (unsigned)((gaddr >> 32) & 0x1FFFFFFull)        // global_addr[56:32]
          | (2u << 30);                                     // type = 2 (image)
    vi8 g1;
    g1[0] = (2 << 16)        // data_size = 4B
          | (1 << 20)        // pad_enable
          | (5 << 22)        // pad_interval: every 64 dwords
          | (1 << 25);       // pad_amount: 2 dwords
    g1[1] = (NLAT & 0xFFFF) << 16;          // tensor_dim0[15:0] = 512 (k)
    g1[2] = (NLAT & 0xFFFF) << 16;          // tensor_dim1[15:0] = 512 (l)
    g1[3] = 64 << 16;                       // tile_dim0 = 64 (k)
    g1[4] = 16;                             // tile_dim1 = 16 (l), tile_dim2 = 0
    g1[5] = NLAT;                           // tensor_dim0_stride = 512
    g1[6] = 0;
    g1[7] = 0;
    vi4 z4 = {0, 0, 0, 0};
    vi8 z8 = {0, 0, 0, 0, 0, 0, 0, 0};
    __builtin_amdgcn_tensor_load_to_lds(g0, g1, z4, z4, z8, 0);
}

// ---------------------------------------------------------------------------
// Stage 3: per-mode Legendre GEMM.
//   out[bc,l,m] = sum_k X[m][bc,k] * W[m][l,k]   (r and i share B fragments)
// Block = 4 waves covering all 64 bc rows for one (m, l-tile).  The weights
// tile streams via the Tensor Data Mover into a double-buffered LDS slot,
// overlapped with WMMA compute; wave 0 owns the TENSORcnt.
// ---------------------------------------------------------------------------
__global__ void sht_leg_gemm(const float* __restrict__ w,
                             const float* __restrict__ xr,
                             const float* __restrict__ xi,
                             float* __restrict__ out) {
    __shared__ float bs[2][TILE_DW];

    int m    = blockIdx.y;
    int l0   = blockIdx.x << 4;
    int wave = threadIdx.x >> 5;
    int lane = threadIdx.x & 31;
    int bc0  = wave << 4;
    int l16  = lane & 15;
    int kb   = (lane >> 4) << 1;

    const float* arp = xr + (size_t)m * RROWS + (size_t)(bc0 + l16) * NLAT + kb;
    const float* aip = xi + (size_t)m * RROWS + (size_t)(bc0 + l16) * NLAT + kb;
    const unsigned long long wtile0 =
        (unsigned long long)(const char*)(w + (size_t)m * NLAT * NLAT
                                            + (size_t)l0 * NLAT);
    unsigned lds0 = (unsigned)(size_t)(&bs[0][0]);

    vf8 accr = {};
    vf8 acci = {};

    if (wave == 0)
        tdm_load_tile(lds0, wtile0);          // k-chunk 0 -> buffer 0

    int buf = 0;
    for (int k0 = 0; k0 < NLAT; k0 += 64) {
        if (wave == 0) {
            if (k0 + 64 < NLAT) {
                tdm_load_tile(lds0 + (unsigned)((buf ^ 1) * (TILE_DW * 4)),
                              wtile0 + (unsigned long long)(k0 + 64) * 4u);
                __builtin_amdgcn_s_wait_tensorcnt(1);   // chunk k0 landed
            } else {
                __builtin_amdgcn_s_wait_tensorcnt(0);   // last chunk landed
            }
        }
        __syncthreads();

        const float* bp = &bs[buf][0] + (size_t)l16 * TROW + kb;
        #pragma unroll
        for (int kk = 0; kk < 64; kk += 4) {
            vf2 ar = *(const vf2*)(arp + k0 + kk);
            vf2 ai = *(const vf2*)(aip + k0 + kk);
            vf2 b  = *(const vf2*)(bp + kk);
            accr = __builtin_amdgcn_wmma_f32_16x16x4_f32(false, ar, false, b,
                                                         (short)0, accr, false, false);
            acci = __builtin_amdgcn_wmma_f32_16x16x4_f32(false, ai, false, b,
                                                         (short)0, acci, false, false);
        }
        __syncthreads();
        buf ^= 1;
    }

    int bcbase = bc0 + ((lane >> 4) << 3);
    int l = l0 + l16;
    #pragma unroll
    for (int v = 0; v < 8; ++v) {
        int bc = bcbase + v;
        size_t oidx = (((size_t)(bc * NLAT + l)) * MMAX + m) << 1;
        vf2 o;
        o[0] = accr[v];
        o[1] = acci[v];
        *(vf2*)(out + oidx) = o;
    }
}

// ---------------------------------------------------------------------------
extern "C" void kernel_launch(void* const* d_in, const int* in_sizes, int n_in,
                              void* d_out, int out_size, void* d_ws, size_t ws_size,
                              hipStream_t stream) {
    const float* x = (const float*)d_in[0];   // (2,32,512,1024) f32
    const float* w = (const float*)d_in[1];   // (513,512,512) f32
    float* out = (float*)d_out;               // (2,32,512,513) complex64 interleaved

    float* fr = (float*)d_ws;                         // MP x NLON
    float* fi = fr + (size_t)TABN;                    // MP x NLON
    float* xr = fi + (size_t)TABN;                    // MMAX x RROWS
    float* xi = xr + (size_t)MMAX * RROWS;            // MMAX x RROWS

    // Stage 1: twiddle tables
    sht_build_dft<<<(TABN + 255) / 256, 256, 0, stream>>>(fr, fi);

    // Stage 2: DFT GEMM — 1024 row-supertiles x 33 mode-tiles, 8 waves/block
    int tiles = (RROWS / 32) * (MP / 16);     // 33792
    sht_dft_gemm<<<tiles / 8, 256, 0, stream>>>(x, fr, fi, xr, xi);

    // Stage 3: Legendre GEMM — grid (l-tiles, modes), 4 waves/block, TDM-fed
    dim3 grid(NLAT / 16, MMAX);
    sht_leg_gemm<<<grid, 128, 0, stream>>>(w, xr, xi, out);
}